// PureSSM_27341761806936
// MI455X (gfx1250) — compile-verified
//
#include <hip/hip_runtime.h>
#include <hip/hip_bf16.h>
#include <math.h>

// ---------------------------------------------------------------------------
// Mamba2-style SSM forward for MI455X (gfx1250, wave32).
// Pipeline: [WMMA fp32 GEMM proj] -> [conv4+SiLU] -> [per-head serial scan]
//           -> [WMMA fp32 GEMM out]
// GEMMs use v_wmma_f32_16x16x4_f32 with double-buffered LDS staged by
// GLOBAL_LOAD_ASYNC_TO_LDS_B128 (ASYNCcnt-gated) when the toolchain exposes
// the async builtins; otherwise falls back to synchronous staging.
// ---------------------------------------------------------------------------

typedef __attribute__((ext_vector_type(2))) float v2f;
typedef __attribute__((ext_vector_type(8))) float v8f;
typedef int v4i __attribute__((vector_size(16)));   // matches builtin proto

#define AS1 __attribute__((address_space(1)))
#define AS3 __attribute__((address_space(3)))

#if defined(__has_builtin)
#  if __has_builtin(__builtin_amdgcn_global_load_async_to_lds_b128) && \
      __has_builtin(__builtin_amdgcn_s_wait_asynccnt)
#    define USE_ASYNC_LDS 1
#  endif
#endif
#ifndef USE_ASYNC_LDS
#  define USE_ASYNC_LDS 0
#endif

#if USE_ASYNC_LDS
#  define WAIT_ASYNC() __builtin_amdgcn_s_wait_asynccnt(0)
#else
#  define WAIT_ASYNC() ((void)0)
#endif

#define D_MODEL   1024
#define D_STATE   64
#define D_CONV    4
#define D_INNER   2048
#define NHEADS    32
#define HEADDIM   64
#define D_PROJ    8224          // 2*D_INNER + NHEADS*(2*D_STATE+1)
#define SEQLEN    4096
#define BATCH     2
#define MTOT      (BATCH * SEQLEN)

// ------------------------- WMMA fp32 GEMM (C = A * B^T) ---------------------
// A[M,K] row-major, B[N,K] row-major (both K-contiguous), C[M,N].
// Block tile 128x128, 256 threads = 8 waves in a 4(M) x 2(N) grid,
// wave tile 32x64 = 2x4 tiles of 16x16. K double-buffered in chunks of 32.
#define BM 128
#define BN 128
#define BK 32
#define LDP (BK + 4)     // pad to 36 floats: keeps 16B alignment for B128 LDS
                         // writes; 36 mod 64 banks -> conflict-free frag reads

__global__ __launch_bounds__(256)
void ssm_wmma_gemm_nt(const float* __restrict__ A,
                      const float* __restrict__ B,
                      float* __restrict__ C,
                      int M, int N, int K) {
  __shared__ float As[2][BM * LDP];
  __shared__ float Bs[2][BN * LDP];

  const int tid      = threadIdx.x;
  const int wave     = tid >> 5;
  const int lane     = tid & 31;
  const int laneHalf = lane >> 4;     // 0: lanes 0-15, 1: lanes 16-31
  const int lane16   = lane & 15;
  const int waveM    = wave >> 1;     // 0..3
  const int waveN    = wave & 1;      // 0..1
  const int blockM   = blockIdx.y * BM;
  const int blockN   = blockIdx.x * BN;

  v8f acc[2][4];
#pragma unroll
  for (int mi = 0; mi < 2; ++mi)
#pragma unroll
    for (int ni = 0; ni < 4; ++ni)
      acc[mi][ni] = (v8f){0.f, 0.f, 0.f, 0.f, 0.f, 0.f, 0.f, 0.f};

  // Cooperative staging: 2 threads per row, 16 contiguous floats each.
  const int ldRow = tid >> 1;          // 0..127
  const int ldCol = (tid & 1) * 16;    // 0 or 16
  const float* aRow  = A + (size_t)(blockM + ldRow) * K + ldCol;
  const int    gn    = blockN + ldRow;
  const bool   bOk   = (gn < N);
  const float* bRow  = B + (size_t)(bOk ? gn : 0) * K + ldCol;
  const int    lBase = ldRow * LDP + ldCol;

  // Issue one K-stage of global->LDS staging into buffer `sb`.
  auto issue = [&](int sb, int k0) {
#if USE_ASYNC_LDS
#pragma unroll
    for (int v = 0; v < 4; ++v)
      __builtin_amdgcn_global_load_async_to_lds_b128(
          (AS1 v4i*)(aRow + k0 + v * 4),
          (AS3 v4i*)(&As[sb][lBase + v * 4]),
          /*offset=*/0, /*cpol=*/0);
    if (bOk) {
#pragma unroll
      for (int v = 0; v < 4; ++v)
        __builtin_amdgcn_global_load_async_to_lds_b128(
            (AS1 v4i*)(bRow + k0 + v * 4),
            (AS3 v4i*)(&Bs[sb][lBase + v * 4]),
            /*offset=*/0, /*cpol=*/0);
    } else {
#pragma unroll
      for (int v = 0; v < 16; ++v) Bs[sb][lBase + v] = 0.f;
    }
#else
#pragma unroll
    for (int v = 0; v < 4; ++v) {
      float4 q = reinterpret_cast<const float4*>(aRow + k0)[v];
      As[sb][lBase + v * 4 + 0] = q.x; As[sb][lBase + v * 4 + 1] = q.y;
      As[sb][lBase + v * 4 + 2] = q.z; As[sb][lBase + v * 4 + 3] = q.w;
    }
    if (bOk) {
#pragma unroll
      for (int v = 0; v < 4; ++v) {
        float4 q = reinterpret_cast<const float4*>(bRow + k0)[v];
        Bs[sb][lBase + v * 4 + 0] = q.x; Bs[sb][lBase + v * 4 + 1] = q.y;
        Bs[sb][lBase + v * 4 + 2] = q.z; Bs[sb][lBase + v * 4 + 3] = q.w;
      }
    } else {
#pragma unroll
      for (int v = 0; v < 16; ++v) Bs[sb][lBase + v] = 0.f;
    }
#endif
  };

  // ---- prologue: stage 0
  issue(0, 0);

  int s = 0;
  for (int k0 = 0; k0 < K; k0 += BK, s ^= 1) {
    WAIT_ASYNC();          // my async writes into buffer s have landed
    __syncthreads();       // everyone's writes landed; everyone done reading s^1

    if (k0 + BK < K) issue(s ^ 1, k0 + BK);   // overlap fetch of next stage

    const float* Ac = &As[s][0];
    const float* Bc = &Bs[s][0];

    // ---- 8 WMMA K-steps of 4 per stage
#pragma unroll
    for (int kk = 0; kk < BK; kk += 4) {
      const int kidx = kk + 2 * laneHalf;  // lanes 0-15: K={kk,kk+1}; 16-31: +2

      v2f afrag[2];
#pragma unroll
      for (int mi = 0; mi < 2; ++mi) {
        const int ml = waveM * 32 + mi * 16 + lane16;
        afrag[mi] = (v2f){Ac[ml * LDP + kidx], Ac[ml * LDP + kidx + 1]};
      }
      v2f bfrag[4];
#pragma unroll
      for (int ni = 0; ni < 4; ++ni) {
        const int nl = waveN * 64 + ni * 16 + lane16;
        bfrag[ni] = (v2f){Bc[nl * LDP + kidx], Bc[nl * LDP + kidx + 1]};
      }
#pragma unroll
      for (int mi = 0; mi < 2; ++mi)
#pragma unroll
        for (int ni = 0; ni < 4; ++ni)
          acc[mi][ni] = __builtin_amdgcn_wmma_f32_16x16x4_f32(
              /*neg_a=*/false, afrag[mi],
              /*neg_b=*/false, bfrag[ni],
              /*c_mod=*/(short)0, acc[mi][ni],
              /*reuse_a=*/false, /*reuse_b=*/false);
    }
  }

  // ---- store C (16x16 f32 D layout: VGPR r -> M=r + 8*laneHalf, N=lane16)
#pragma unroll
  for (int mi = 0; mi < 2; ++mi) {
#pragma unroll
    for (int ni = 0; ni < 4; ++ni) {
      const int gnn = blockN + waveN * 64 + ni * 16 + lane16;
      if (gnn < N) {
#pragma unroll
        for (int r = 0; r < 8; ++r) {
          const int gm = blockM + waveM * 32 + mi * 16 + r + 8 * laneHalf;
          C[(size_t)gm * N + gnn] = acc[mi][ni][r];
        }
      }
    }
  }
}

// ------------------------- conv(4) + SiLU ----------------------------------
__global__ __launch_bounds__(256)
void ssm_conv_silu(const float* __restrict__ proj,
                   const float* __restrict__ cw,
                   const float* __restrict__ cb,
                   float* __restrict__ xc) {
  const int i = blockIdx.x * blockDim.x + threadIdx.x;   // over MTOT*D_INNER
  const int c = i % D_INNER;
  const int m = i / D_INNER;
  const int l = m % SEQLEN;

  float acc = cb[c];
#pragma unroll
  for (int k = 0; k < D_CONV; ++k) {
    const int lt = l + k - (D_CONV - 1);
    if (lt >= 0)
      acc += cw[c * D_CONV + k] * proj[(size_t)(m + k - (D_CONV - 1)) * D_PROJ + c];
  }
  xc[(size_t)m * D_INNER + c] = acc * (1.f / (1.f + __expf(-acc)));   // SiLU
}

// ------------------------- selective scan ----------------------------------
// One workgroup per (batch, head). 256 threads hold the full 64x64 state:
// thread (g = tid>>6, d = tid&63) owns h[s = g*16 .. g*16+15][d].
__device__ __forceinline__ float softplus_f(float x) {
  return (x > 20.f) ? x : log1pf(__expf(x));
}

__global__ __launch_bounds__(256)
void ssm_scan(const float* __restrict__ proj,
              const float* __restrict__ xc,
              const float* __restrict__ A_log,
              const float* __restrict__ Dp,
              const float* __restrict__ dt_bias,
              float* __restrict__ yg) {
  const int b = blockIdx.x / NHEADS;
  const int n = blockIdx.x % NHEADS;
  const int tid = threadIdx.x;
  const int d = tid & 63;
  const int g = tid >> 6;          // 0..3
  const int sBase = g * 16;

  __shared__ float sB[D_STATE];
  __shared__ float sC[D_STATE];
  __shared__ float sX[HEADDIM];
  __shared__ float sY[256];

  const float dtb = dt_bias[n];
  const float Dn  = Dp[n];

  float Aneg[16];
#pragma unroll
  for (int j = 0; j < 16; ++j)
    Aneg[j] = -__expf(A_log[n * D_STATE + sBase + j]);

  float h[16];
#pragma unroll
  for (int j = 0; j < 16; ++j) h[j] = 0.f;

  const size_t headOff = (size_t)2 * D_INNER + (size_t)n * (2 * D_STATE + 1);

  for (int t = 0; t < SEQLEN; ++t) {
    const int m = b * SEQLEN + t;
    const size_t pRow = (size_t)m * D_PROJ;

    // stage B_proj, C_proj, x_heads for this timestep
    if (tid < 64)        sB[tid]       = proj[pRow + headOff + tid];
    else if (tid < 128)  sC[tid - 64]  = proj[pRow + headOff + D_STATE + (tid - 64)];
    else if (tid < 192)  sX[tid - 128] = xc[(size_t)m * D_INNER + n * HEADDIM + (tid - 128)];
    __syncthreads();

    const float dtraw = proj[pRow + headOff + 2 * D_STATE];  // uniform -> scalar load
    const float dtv   = softplus_f(dtraw + dtb);
    const float xd    = sX[d];

    float partial = 0.f;
#pragma unroll
    for (int j = 0; j < 16; ++j) {
      const int s = sBase + j;
      const float dA = __expf(Aneg[j] * dtv);
      h[j] = dA * h[j] + (sB[s] * dtv) * xd;
      partial += sC[s] * h[j];
    }
    sY[tid] = partial;
    __syncthreads();

    if (tid < 64) {
      const float ysum = sY[tid] + sY[tid + 64] + sY[tid + 128] + sY[tid + 192]
                       + Dn * sX[tid];
      const float zv = proj[pRow + D_INNER + n * HEADDIM + tid];
      const float gate = zv * (1.f / (1.f + __expf(-zv)));   // SiLU(z)
      yg[(size_t)m * D_INNER + n * HEADDIM + tid] = ysum * gate;
    }
    __syncthreads();   // protect shared staging before next timestep
  }
}

// ------------------------- launcher ----------------------------------------
extern "C" void kernel_launch(void* const* d_in, const int* in_sizes, int n_in,
                              void* d_out, int out_size, void* d_ws, size_t ws_size,
                              hipStream_t stream) {
  const float* x      = (const float*)d_in[0];
  const float* W_in   = (const float*)d_in[1];
  const float* conv_w = (const float*)d_in[2];
  const float* conv_b = (const float*)d_in[3];
  const float* A_log  = (const float*)d_in[4];
  const float* Dvec   = (const float*)d_in[5];
  const float* dt_b   = (const float*)d_in[6];
  const float* W_out  = (const float*)d_in[7];
  float* out = (float*)d_out;

  char* ws = (char*)d_ws;
  float* proj = (float*)ws;                                             // 8192 x 8224
  float* xc   = (float*)(ws + (size_t)MTOT * D_PROJ * sizeof(float));   // 8192 x 2048
  float* yg   = (float*)(ws + (size_t)MTOT * D_PROJ * sizeof(float)
                            + (size_t)MTOT * D_INNER * sizeof(float));  // 8192 x 2048

  // 1) input projection: proj = x * W_in^T   (M=8192, N=8224, K=1024)
  {
    dim3 grid((D_PROJ + BN - 1) / BN, MTOT / BM);
    ssm_wmma_gemm_nt<<<grid, 256, 0, stream>>>(x, W_in, proj, MTOT, D_PROJ, D_MODEL);
  }
  // 2) depthwise conv + SiLU
  ssm_conv_silu<<<(MTOT * D_INNER) / 256, 256, 0, stream>>>(proj, conv_w, conv_b, xc);
  // 3) selective scan + skip + gating
  ssm_scan<<<BATCH * NHEADS, 256, 0, stream>>>(proj, xc, A_log, Dvec, dt_b, yg);
  // 4) output projection: out = yg * W_out^T  (M=8192, N=1024, K=2048)
  {
    dim3 grid(D_MODEL / BN, MTOT / BM);
    ssm_wmma_gemm_nt<<<grid, 256, 0, stream>>>(yg, W_out, out, MTOT, D_MODEL, D_INNER);
  }
}